// VectorQuantizer_87093346828947
// MI455X (gfx1250) — compile-verified
//
#include <hip/hip_runtime.h>
#include <math.h>
#include <stdint.h>

typedef float v2f __attribute__((ext_vector_type(2)));
typedef float v8f __attribute__((ext_vector_type(8)));
typedef unsigned int v4u __attribute__((ext_vector_type(4)));
typedef int v4i __attribute__((ext_vector_type(4)));
typedef int v8i __attribute__((ext_vector_type(8)));

#define N_CODES  1024
#define CODE_DIM 64
#define NVEC     32768          // 32*32*32 vectors
#define NELEM    2097152        // 32*64*32*32 elements

// LDS tile row stride: 64 data floats + 4 pad floats (TDM pad feature) so that
// ds_load_b64 B-fragment reads are bank-conflict-free (64 banks).
#define TILE_ROW_F 68

// workspace layout (floats)
#define WS_ENORM  0             // 1024: ||e_k||^2
#define WS_COUNTS 1024          // 1024: usage counts
#define WS_SUMSQ  2048          // 1:    sum (z_q - z_e)^2

// d_out layout (floats), reference tuple order
#define OUT_ZQ   0              // 2097152
#define OUT_LOSS 2097152        // 1
#define OUT_IDX  2097153        // 32768 (indices as float)
#define OUT_PERP 2129921        // 1
#define OUT_PROB 2129922        // 1024

// ---------------------------------------------------------------------------
// TDM: DMA a 16-code x 64-float codebook tile into LDS, with 4-DWORD padding
// inserted after every 64 DWORDs (-> LDS row stride 68 floats).
// D# per CDNA5 ISA ch.8: group0 = {count/lds_addr/global_addr/type},
// group1 = {data_size, pad cfg, tensor dims 64x1024, tile 64x16, stride 64}.
__device__ __forceinline__ void tdm_load_tile(const float* cb, int codeBase,
                                              uint32_t lds_off) {
    uint64_t gaddr = (uint64_t)(uintptr_t)(cb + (size_t)codeBase * CODE_DIM);
    v4u g0;
    g0.x = 1u;                                  // count=1, user descriptor
    g0.y = lds_off;                             // lds_addr (bytes)
    g0.z = (uint32_t)gaddr;                     // global_addr[31:0]
    g0.w = (uint32_t)((gaddr >> 32) & 0x1FFFFFFu) | (2u << 30); // [56:32]+type=2
    v8i g1;
    g1[0] = (int)((2u << 16)        // data_size = 4 bytes
                | (1u << 20)        // pad_enable
                | (5u << 22)        // pad_interval: 64 DWORDs
                | (3u << 25));      // pad_amount: 4 DWORDs
    g1[1] = (int)(64u << 16);       // tensor_dim0 = 64 (low16 at bits 63:48)
    g1[2] = (int)(1024u << 16);     // tensor_dim1 = 1024 (low16 at bits 95:80)
    g1[3] = (int)(64u << 16);       // tile_dim0 = 64 (bits 127:112)
    g1[4] = 16;                     // tile_dim1 = 16, tile_dim2 = 0
    g1[5] = 64;                     // tensor_dim0_stride = 64
    g1[6] = 0;
    g1[7] = 0;
    v4i gz = {0, 0, 0, 0};
#if defined(__clang_major__) && (__clang_major__ >= 23)
    v8i gz8 = {0, 0, 0, 0, 0, 0, 0, 0};
    __builtin_amdgcn_tensor_load_to_lds(g0, g1, gz, gz, gz8, 0);
#else
    __builtin_amdgcn_tensor_load_to_lds(g0, g1, gz, gz, 0);
#endif
}

// ---------------------------------------------------------------------------
// K0: precompute codebook norms, zero accumulators
__global__ void vq_init(const float* __restrict__ cb, float* __restrict__ ws) {
    int k = blockIdx.x * blockDim.x + threadIdx.x;
    if (k < N_CODES) {
        const float* row = cb + k * CODE_DIM;
        float s = 0.f;
#pragma unroll
        for (int d = 0; d < CODE_DIM; d += 2) {
            float2 v = *(const float2*)(row + d);
            s += v.x * v.x + v.y * v.y;
        }
        ws[WS_ENORM + k]  = s;
        ws[WS_COUNTS + k] = 0.f;
        if (k == 0) ws[WS_SUMSQ] = 0.f;
    }
}

// ---------------------------------------------------------------------------
// K1: WMMA distance matmul + argmin. One wave handles 16 z-rows vs all codes.
// Codebook tiles are TDM-DMA'd into double-buffered LDS by wave 0 and shared
// by all 4 waves of the block (4x less L2 traffic, DMA overlapped with WMMA).
// A frag (16x4 f32): lane L -> M = L%16, K = 2*(L/16) + {0,1}
// B frag (4x16 f32): lane L -> N = L%16, K = 2*(L/16) + {0,1}
// C/D (16x16 f32):   lane L, vgpr r -> (row r + 8*(L/16), col L%16)
__global__ __launch_bounds__(128)
void vq_argmin_wmma(const float* __restrict__ z, const float* __restrict__ cb,
                    const float* __restrict__ enorm, float* __restrict__ out_idx) {
    __shared__ float lds_buf[2][16 * TILE_ROW_F];

    const int lane = threadIdx.x & 31;
    const int wave = threadIdx.x >> 5;
    const int rowBase = (blockIdx.x * 4 + wave) * 16;
    const int m  = lane & 15;          // row (A) / col (B) within tile
    const int kh = (lane >> 4) << 1;   // K sub-offset: 0 or 2

    const uint32_t lds_off0 = (uint32_t)(uintptr_t)&lds_buf[0][0];
    const uint32_t lds_off1 = (uint32_t)(uintptr_t)&lds_buf[1][0];

    // kick off DMA of code tile 0 while we load A fragments
    if (wave == 0) tdm_load_tile(cb, 0, lds_off0);

    // z is (B,D,H,W); vector n = b*1024 + h*32 + w, element d strides by 1024
    const int n  = rowBase + m;
    const int b  = n >> 10;
    const int hw = n & 1023;
    const float* zrow = z + (size_t)b * 65536 + hw;

    v2f a[16];
#pragma unroll
    for (int s = 0; s < 16; ++s) {
        int d = 4 * s + kh;
        a[s].x = zrow[(size_t)d * 1024];
        a[s].y = zrow[(size_t)(d + 1) * 1024];
    }

    float bestD[8];
    int   bestI[8];
#pragma unroll
    for (int r = 0; r < 8; ++r) { bestD[r] = 3.4e38f; bestI[r] = 0; }

    __builtin_amdgcn_s_wait_tensorcnt(0);   // no-op for waves 1..3
    __syncthreads();

    for (int kt = 0; kt < 64; ++kt) {        // 64 tiles of 16 codes
        const int cur = kt & 1;
        if (wave == 0 && (kt + 1) < 64)      // prefetch next tile via TDM
            tdm_load_tile(cb, (kt + 1) * 16, cur ? lds_off0 : lds_off1);

        const float* brow = &lds_buf[cur][m * TILE_ROW_F + kh];
        v2f bf[16];
#pragma unroll
        for (int s = 0; s < 16; ++s) {       // ds_load_b64, conflict-free
            float2 v = *(const float2*)(brow + 4 * s);
            bf[s].x = v.x; bf[s].y = v.y;
        }

        v8f acc = {0.f, 0.f, 0.f, 0.f, 0.f, 0.f, 0.f, 0.f};
#pragma unroll
        for (int s = 0; s < 16; ++s) {
            acc = __builtin_amdgcn_wmma_f32_16x16x4_f32(
                false, a[s], false, bf[s], (short)0, acc, false, false);
        }

        const float en = enorm[kt * 16 + m];
#pragma unroll
        for (int r = 0; r < 8; ++r) {
            float dist = en - 2.0f * acc[r];   // ||e||^2 - 2 z.e (||z||^2 const)
            if (dist < bestD[r]) { bestD[r] = dist; bestI[r] = kt * 16 + m; }
        }

        __builtin_amdgcn_s_wait_tensorcnt(0);  // tile kt+1 DMA done (wave 0)
        __syncthreads();                       // publish LDS to all waves
    }

    // min-reduce across the 16 lanes of each half-wave (wave32)
#pragma unroll
    for (int r = 0; r < 8; ++r) {
        float d = bestD[r]; int i = bestI[r];
#pragma unroll
        for (int off = 8; off >= 1; off >>= 1) {
            float od = __shfl_xor(d, off, 16);
            int   oi = __shfl_xor(i, off, 16);
            if (od < d || (od == d && oi < i)) { d = od; i = oi; }
        }
        if (m == 0) {
            int row = r + ((lane >> 4) << 3);
            out_idx[rowBase + row] = (float)i;
        }
    }
}

// ---------------------------------------------------------------------------
// K2: gather z_q (== z_q_st), accumulate squared error, usage counts
__global__ __launch_bounds__(256)
void vq_gather(const float* __restrict__ z, const float* __restrict__ cb,
               const float* __restrict__ idxf, float* __restrict__ out_zq,
               float* __restrict__ ws) {
    __shared__ float red[256];
    int i = blockIdx.x * 256 + threadIdx.x;    // flat (b,d,h,w)
    int d = (i >> 10) & 63;
    int b = i >> 16;
    int n = (b << 10) | (i & 1023);
    int idx = (int)idxf[n];
    float zq = cb[idx * CODE_DIM + d];
    float ze = z[i];
    out_zq[i] = zq;                            // straight-through value == z_q
    float diff = zq - ze;
    red[threadIdx.x] = diff * diff;
    if (d == 0) atomicAdd(&ws[WS_COUNTS + idx], 1.0f);
    __syncthreads();
    for (int s = 128; s > 0; s >>= 1) {
        if (threadIdx.x < s) red[threadIdx.x] += red[threadIdx.x + s];
        __syncthreads();
    }
    if (threadIdx.x == 0) atomicAdd(&ws[WS_SUMSQ], red[0]);
}

// ---------------------------------------------------------------------------
// K3: usage probs, perplexity, vq_loss (= 1.25 * MSE, since both losses equal)
__global__ __launch_bounds__(1024)
void vq_finalize(const float* __restrict__ ws, float* __restrict__ out) {
    __shared__ float red[1024];
    int k = threadIdx.x;
    float p = ws[WS_COUNTS + k] * (1.0f / (float)NVEC);
    out[OUT_PROB + k] = p;
    red[k] = p * logf(p + 1e-10f);
    __syncthreads();
    for (int s = 512; s > 0; s >>= 1) {
        if (k < s) red[k] += red[k + s];
        __syncthreads();
    }
    if (k == 0) {
        out[OUT_PERP] = expf(-red[0]);
        out[OUT_LOSS] = 1.25f * ws[WS_SUMSQ] * (1.0f / (float)NELEM);
    }
}

// ---------------------------------------------------------------------------
extern "C" void kernel_launch(void* const* d_in, const int* in_sizes, int n_in,
                              void* d_out, int out_size, void* d_ws, size_t ws_size,
                              hipStream_t stream) {
    const float* z  = (const float*)d_in[0];   // (32,64,32,32)
    const float* cb = (const float*)d_in[1];   // (1024,64)
    float* out = (float*)d_out;
    float* ws  = (float*)d_ws;

    vq_init<<<4, 256, 0, stream>>>(cb, ws);
    vq_argmin_wmma<<<512, 128, 0, stream>>>(z, cb, ws + WS_ENORM, out + OUT_IDX);
    vq_gather<<<NELEM / 256, 256, 0, stream>>>(z, cb, out + OUT_IDX, out + OUT_ZQ, ws);
    vq_finalize<<<1, 1024, 0, stream>>>(ws, out);
}